// Model_Beam_par_16527034155798
// MI455X (gfx1250) — compile-verified
//
#include <hip/hip_runtime.h>

// Radon sinogram for MI455X (gfx1250).
// out[r*nang + a] = sum_c bilinear(padded(725x725), rotate_inv(c, r, angle[a]))
// Strategy: per-block (angle, 64-row) tile, loop over 12 c-tiles of 64.
// For each (r,c)-tile the sample footprint is a rotated box <= 92x92 floats;
// stage it into LDS (async-to-LDS: global_load_async_to_lds_b32), then do all
// bilinear gathers from LDS (ds_load) instead of scattered global loads.
// Final 4-way partial-sum reduction per row is done with V_WMMA_F32_16X16X4_F32
// against an all-ones B matrix (row-sum via matrix engine).

#define DIAG   725          // ceil(sqrt(512^2+512^2))
#define IN_W   512
#define OFFP   106          // (725-512)//2 : data region [106, 617]
#define CTRP   362          // 725//2
#define CT     64           // columns (c) per tile
#define RT     64           // output rows (r) per block
#define TPB    256
#define TILE_MAX 9216       // >= 92*92 worst-case staged box

typedef float v2f __attribute__((ext_vector_type(2)));
typedef float v8f __attribute__((ext_vector_type(8)));

#if defined(__has_builtin)
#  if __has_builtin(__builtin_amdgcn_global_load_async_to_lds_b32)
#    define USE_ASYNC_LDS 1
#  endif
#endif

#ifdef USE_ASYNC_LDS
typedef __attribute__((address_space(1))) int g1_int;   // global (AS1)
typedef __attribute__((address_space(3))) int l3_int;   // LDS (AS3)

__device__ __forceinline__ void wait_async_zero() {
#  if __has_builtin(__builtin_amdgcn_s_wait_asynccnt)
  __builtin_amdgcn_s_wait_asynccnt(0);
#  else
  asm volatile("s_wait_asynccnt 0" ::: "memory");
#  endif
}
#endif

__global__ __launch_bounds__(TPB)
void radon_sinogram_kernel(const float* __restrict__ img,
                           const float* __restrict__ angles,
                           float* __restrict__ out,
                           int nang)
{
  __shared__ float tile[TILE_MAX];
  __shared__ float red[TPB];

  const int tid  = threadIdx.x;
  const int aidx = blockIdx.y;
  const int r0   = blockIdx.x * RT;

  // angle_rad = -deg * pi/180 ; same trig as reference
  const float rad = -angles[aidx] * 0.01745329251994329577f;
  const float ct = cosf(rad);
  const float st = sinf(rad);

  const int   rlane = tid & (RT - 1);        // 0..63  (wave-contiguous)
  const int   cg    = tid >> 6;              // 0..3   (uniform per wave)
  const float fy    = (float)(r0 + rlane - CTRP);
  const float xr_row = -st * fy + (float)CTRP;   // x_rot = ct*fx + xr_row
  const float yr_row =  ct * fy + (float)CTRP;   // y_rot = st*fx + yr_row

  const int stg_col  = tid & 127;            // staging: 128 cols x 2 rows/pass
  const int stg_row0 = tid >> 7;             // 0..1

  float acc = 0.0f;

  const int n_ctiles = (DIAG + CT - 1) / CT; // 12
  for (int t = 0; t < n_ctiles; ++t) {
    const int c0 = t * CT;

    // Bounding box of sample points over tile corners (affine => corner extrema).
    const float fxa = (float)(c0 - CTRP);
    const float fxb = (float)(c0 + CT - 1 - CTRP);
    const float fya = (float)(r0 - CTRP);
    const float fyb = (float)(r0 + RT - 1 - CTRP);

    const float xmn = (float)CTRP + fminf(ct*fxa, ct*fxb) + fminf(-st*fya, -st*fyb);
    const float xmx = (float)CTRP + fmaxf(ct*fxa, ct*fxb) + fmaxf(-st*fya, -st*fyb);
    const float ymn = (float)CTRP + fminf(st*fxa, st*fxb) + fminf(ct*fya, ct*fyb);
    const float ymx = (float)CTRP + fmaxf(st*fxa, st*fxb) + fmaxf(ct*fya, ct*fyb);

    // [clamp(floor(min)), clamp(floor(max)+1)] covers every clipped tap index.
    const int xmin_i = min(max((int)floorf(xmn),     0), DIAG-1);
    const int xmax_i = min(max((int)floorf(xmx) + 1, 0), DIAG-1);
    const int ymin_i = min(max((int)floorf(ymn),     0), DIAG-1);
    const int ymax_i = min(max((int)floorf(ymx) + 1, 0), DIAG-1);
    const int wb = xmax_i - xmin_i + 1;      // <= 92
    const int hb = ymax_i - ymin_i + 1;      // <= 92

    __syncthreads();                         // previous tile fully consumed

    // Stage conceptual-padded [ymin..ymax] x [xmin..xmax] into LDS.
    for (int row = stg_row0; row < hb; row += 2) {
      if (stg_col < wb) {
        const int gy   = ymin_i + row     - OFFP;   // input coords
        const int gx   = xmin_i + stg_col - OFFP;
        const int lidx = row * wb + stg_col;
        const bool inr = ((unsigned)gy < (unsigned)IN_W) &&
                         ((unsigned)gx < (unsigned)IN_W);
#ifdef USE_ASYNC_LDS
        if (inr) {
          __builtin_amdgcn_global_load_async_to_lds_b32(
              (g1_int*)(img + (gy * IN_W + gx)),
              (l3_int*)&tile[lidx], 0, 0);
        } else {
          tile[lidx] = 0.0f;
        }
#else
        tile[lidx] = inr ? img[gy * IN_W + gx] : 0.0f;
#endif
      }
    }
#ifdef USE_ASYNC_LDS
    wait_async_zero();
#endif
    __syncthreads();

    // 16 c-samples per thread: c = c0 + cg + 4k. All gathers hit LDS.
#pragma unroll 4
    for (int k = 0; k < CT / 4; ++k) {
      const int c = c0 + cg + 4 * k;
      if (c < DIAG) {
        const float fx    = (float)(c - CTRP);
        const float x_rot = ct * fx + xr_row;
        const float y_rot = st * fx + yr_row;
        int x0i = (int)floorf(x_rot);
        int y0i = (int)floorf(y_rot);
        x0i = min(max(x0i, 0), DIAG - 1);
        y0i = min(max(y0i, 0), DIAG - 1);
        const int x1i = min(x0i + 1, DIAG - 1);
        const int y1i = min(y0i + 1, DIAG - 1);
        const float dx1 = (float)x1i - x_rot;
        const float dx0 = x_rot - (float)x0i;
        const float dy1 = (float)y1i - y_rot;
        const float dy0 = y_rot - (float)y0i;
        const int bx0 = x0i - xmin_i;
        const int bx1 = x1i - xmin_i;
        const int by0 = (y0i - ymin_i) * wb;
        const int by1 = (y1i - ymin_i) * wb;
        const float Ia = tile[by0 + bx0];
        const float Ib = tile[by0 + bx1];
        const float Ic = tile[by1 + bx0];
        const float Id = tile[by1 + bx1];
        acc += (dx1 * dy1) * Ia + (dx0 * dy1) * Ib
             + (dx1 * dy0) * Ic + (dx0 * dy0) * Id;
      }
    }
  }

  // red[cg*64 + r] holds the 4 column-group partials of each of the 64 rows.
  red[tid] = acc;
  __syncthreads();

  // Reduce 4 partials per row with V_WMMA_F32_16X16X4_F32 against all-ones B:
  // D[m,n] = sum_k A[m,k]. Waves 0..3 (full waves, EXEC all-1s) each do 16 rows.
  const int wid = tid >> 5;
  if (wid < 4) {
    const int lane = tid & 31;
    const int mrow = (wid << 4) + (lane & 15);   // block-local row handled by this A-row
    const int g0   = (lane < 16) ? 0 : 2;        // A 16x4 layout: lanes 0-15 K={0,1}, 16-31 K={2,3}
    v2f a;
    a.x = red[(g0    ) * RT + mrow];
    a.y = red[(g0 + 1) * RT + mrow];
    v2f b; b.x = 1.0f; b.y = 1.0f;               // all-ones => K-order irrelevant
    v8f cz = {};
    v8f d = __builtin_amdgcn_wmma_f32_16x16x4_f32(
        /*neg_a=*/false, a, /*neg_b=*/false, b,
        /*c_mod=*/(short)0, cz, /*reuse_a=*/false, /*reuse_b=*/false);
    // C/D layout: lane 0 (N=0) holds M=0..7 in VGPR 0..7; lane 16 holds M=8..15.
    if (lane == 0 || lane == 16) {
      const int mbase = (wid << 4) + ((lane == 16) ? 8 : 0);
#pragma unroll
      for (int v = 0; v < 8; ++v) {
        const int rr = r0 + mbase + v;
        if (rr < DIAG) out[rr * nang + aidx] = d[v];
      }
    }
  }
}

extern "C" void kernel_launch(void* const* d_in, const int* in_sizes, int n_in,
                              void* d_out, int out_size, void* d_ws, size_t ws_size,
                              hipStream_t stream) {
  (void)n_in; (void)out_size; (void)d_ws; (void)ws_size;
  const float* x      = (const float*)d_in[0];
  const float* angles = (const float*)d_in[1];
  float* out          = (float*)d_out;
  const int nang = in_sizes[1];                 // 180
  dim3 grid((DIAG + RT - 1) / RT, nang, 1);     // 12 x 180 blocks
  radon_sinogram_kernel<<<grid, dim3(TPB, 1, 1), 0, stream>>>(x, angles, out, nang);
}